// ConvDeepSet_62173946577442
// MI455X (gfx1250) — compile-verified
//
#include <hip/hip_runtime.h>

#define B_SZ    4
#define N_IN    1024
#define N_OUT   1024
#define IN_CH   8      // density + 7 raw channels
#define OUT_CH  16
#define BLOCK   128
#define WAVES   (BLOCK / 32)
#define OUT_TILE (WAVES * 16)   // 64 outputs per block

typedef __attribute__((ext_vector_type(2))) float v2f;
typedef __attribute__((ext_vector_type(8))) float v8f;

__global__ __launch_bounds__(BLOCK)
void convdeepset_kernel(const float* __restrict__ cx,     // (B, N_IN, 1)
                        const float* __restrict__ cy,     // (B, N_IN, 7)
                        const float* __restrict__ t,      // (B, N_OUT, 1)
                        const float* __restrict__ sigma,  // (8,)
                        const float* __restrict__ W,      // (8, 16) row-major
                        const float* __restrict__ bias,   // (16,)
                        float* __restrict__ out)          // (B, N_OUT, 16)
{
    __shared__ __align__(16) float s_x[N_IN];
    __shared__ __align__(16) float s_y[N_IN][IN_CH];   // [1.0, y0..y6]

    const int tid       = threadIdx.x;
    const int tiles_pb  = N_OUT / OUT_TILE;
    const int batch     = blockIdx.x / tiles_pb;
    const int obase_blk = (blockIdx.x % tiles_pb) * OUT_TILE;

    // kick L2 prefetch of this batch's context while we stage it
    __builtin_prefetch(cx + batch * N_IN, 0, 1);
    __builtin_prefetch(cy + batch * N_IN * 7, 0, 1);

    // ---- stage context into LDS (density channel prepended as 1.0) ----
    for (int i = tid; i < N_IN; i += BLOCK) {
        s_x[i]    = cx[batch * N_IN + i];
        s_y[i][0] = 1.0f;
#pragma unroll
        for (int c = 0; c < 7; ++c)
            s_y[i][c + 1] = cy[(batch * N_IN + i) * 7 + c];
    }
    __syncthreads();

    const int wave = tid >> 5;
    const int lane = tid & 31;
    const int row  = lane & 15;   // output row within wave tile; also N column
    const bool hi  = (lane >> 4) != 0;  // which half of the i-range this lane sums

    const int   o  = obase_blk + wave * 16 + row;
    const float tv = t[batch * N_OUT + o];

    // exp(-0.5*d2/s_c^2) = exp2(d2 * m_c),  m_c = -0.5*log2e*exp(-2*sigma_c)
    const float L2E = 1.44269504088896340736f;
    float m0 = -0.5f * L2E * __builtin_amdgcn_exp2f(-2.0f * sigma[0] * L2E);
    float m1 = -0.5f * L2E * __builtin_amdgcn_exp2f(-2.0f * sigma[1] * L2E);
    float m2 = -0.5f * L2E * __builtin_amdgcn_exp2f(-2.0f * sigma[2] * L2E);
    float m3 = -0.5f * L2E * __builtin_amdgcn_exp2f(-2.0f * sigma[3] * L2E);
    float m4 = -0.5f * L2E * __builtin_amdgcn_exp2f(-2.0f * sigma[4] * L2E);
    float m5 = -0.5f * L2E * __builtin_amdgcn_exp2f(-2.0f * sigma[5] * L2E);
    float m6 = -0.5f * L2E * __builtin_amdgcn_exp2f(-2.0f * sigma[6] * L2E);
    float m7 = -0.5f * L2E * __builtin_amdgcn_exp2f(-2.0f * sigma[7] * L2E);

    float a0v = 0.f, a1v = 0.f, a2v = 0.f, a3v = 0.f;
    float a4v = 0.f, a5v = 0.f, a6v = 0.f, a7v = 0.f;

    // ---- main reduction: each lane sums its half of the context points ----
    const int ibeg = hi ? (N_IN / 2) : 0;
#pragma unroll 2
    for (int i = ibeg; i < ibeg + N_IN / 2; ++i) {
        const float dx = s_x[i] - tv;
        const float d2 = dx * dx;
        const float4 ya = *reinterpret_cast<const float4*>(&s_y[i][0]);
        const float4 yb = *reinterpret_cast<const float4*>(&s_y[i][4]);
        a0v = fmaf(__builtin_amdgcn_exp2f(d2 * m0), ya.x, a0v);
        a1v = fmaf(__builtin_amdgcn_exp2f(d2 * m1), ya.y, a1v);
        a2v = fmaf(__builtin_amdgcn_exp2f(d2 * m2), ya.z, a2v);
        a3v = fmaf(__builtin_amdgcn_exp2f(d2 * m3), ya.w, a3v);
        a4v = fmaf(__builtin_amdgcn_exp2f(d2 * m4), yb.x, a4v);
        a5v = fmaf(__builtin_amdgcn_exp2f(d2 * m5), yb.y, a5v);
        a6v = fmaf(__builtin_amdgcn_exp2f(d2 * m6), yb.z, a6v);
        a7v = fmaf(__builtin_amdgcn_exp2f(d2 * m7), yb.w, a7v);
    }

    // ---- combine the two half-range partials (lane L <-> L+16) ----
    const float f0 = a0v + __shfl_xor(a0v, 16, 32);
    const float f1 = a1v + __shfl_xor(a1v, 16, 32);
    const float f2 = a2v + __shfl_xor(a2v, 16, 32);
    const float f3 = a3v + __shfl_xor(a3v, 16, 32);
    const float f4 = a4v + __shfl_xor(a4v, 16, 32);
    const float f5 = a5v + __shfl_xor(a5v, 16, 32);
    const float f6 = a6v + __shfl_xor(a6v, 16, 32);
    const float f7 = a7v + __shfl_xor(a7v, 16, 32);

    const float den  = f0;
    const float rden = 1.0f / (den + 1e-8f);
    const float g1 = f1 * rden, g2 = f2 * rden, g3 = f3 * rden;
    const float g4 = f4 * rden, g5 = f5 * rden, g6 = f6 * rden;
    const float g7 = f7 * rden;

    // ---- epilogue GEMM: (16x8) @ (8x16) + bias via two K=4 f32 WMMAs ----
    // A 16x4 layout: lanes 0-15 -> K=0,1 ; lanes 16-31 -> K=2,3 (VGPR0/1)
    // B 4x16 layout mirrors A: lanes 0-15 -> rows K=0,1 ; lanes 16-31 -> K=2,3
    const int col = row;
    v2f a0, a1, b0, b1;
    a0.x = hi ? g2 : den;          // single v_cndmask each, no select chains
    a0.y = hi ? g3 : g1;
    a1.x = hi ? g6 : g4;
    a1.y = hi ? g7 : g5;

    const int kb = hi ? 2 : 0;     // B-matrix K-row pair base for this lane
    b0.x = W[(kb + 0) * OUT_CH + col];
    b0.y = W[(kb + 1) * OUT_CH + col];
    b1.x = W[(kb + 4) * OUT_CH + col];
    b1.y = W[(kb + 5) * OUT_CH + col];

    v8f d;
    const float bv = bias[col];
#pragma unroll
    for (int r = 0; r < 8; ++r) d[r] = bv;   // C = bias broadcast per column

    d = __builtin_amdgcn_wmma_f32_16x16x4_f32(false, a0, false, b0,
                                              (short)0, d, false, false);
    d = __builtin_amdgcn_wmma_f32_16x16x4_f32(false, a1, false, b1,
                                              (short)0, d, false, false);

    // D layout: VGPR r -> row M = r + 8*half ; lane&15 -> column N
    const int orow_base = obase_blk + wave * 16 + (hi ? 8 : 0);
#pragma unroll
    for (int r = 0; r < 8; ++r)
        out[(batch * N_OUT + orow_base + r) * OUT_CH + col] = d[r];
}

extern "C" void kernel_launch(void* const* d_in, const int* in_sizes, int n_in,
                              void* d_out, int out_size, void* d_ws, size_t ws_size,
                              hipStream_t stream) {
    const float* cx = (const float*)d_in[0];  // context_x
    const float* cy = (const float*)d_in[1];  // context_y
    const float* t  = (const float*)d_in[2];  // t
    const float* sg = (const float*)d_in[3];  // sigma
    const float* W  = (const float*)d_in[4];  // W
    const float* bs = (const float*)d_in[5];  // b
    float* out = (float*)d_out;

    dim3 grid(B_SZ * (N_OUT / OUT_TILE));
    convdeepset_kernel<<<grid, BLOCK, 0, stream>>>(cx, cy, t, sg, W, bs, out);
}